// DifferentiableEKF_56444460204336
// MI455X (gfx1250) — compile-verified
//
#include <hip/hip_runtime.h>
#include <stdint.h>
#include <stddef.h>

namespace {

typedef uint32_t u32x4 __attribute__((ext_vector_type(4)));
typedef uint32_t u32x8 __attribute__((ext_vector_type(8)));

constexpr int kT      = 2048;   // timesteps
constexpr int kTT     = 16;     // timesteps per LDS tile
constexpr int kRows   = 64;     // rows per wave (2 per lane)
constexpr int kRS     = 20;     // LDS row stride in floats (TDM pad: 16DW row + 4DW pad = 80B)
constexpr int kTiles  = kT / kTT;
constexpr int kInSz   = kRows * kRS;     // floats per input slab
constexpr int kBufSz  = 3 * kInSz;       // floats per double-buffer half

// One TDM 2D-tile DMA: 64 rows x 16 floats, row stride kT, +4-DWORD pad per row in LDS.
// D# group0/group1 only (2D tile -> VADDR2/VADDR3 are NULL -> 2-operand form).
__device__ __forceinline__ void tdm_load_2d(uint32_t lds_byte, uint64_t gaddr) {
  u32x4 g0;
  g0.x = 1u;                                       // count=1, user-mode, no gather
  g0.y = lds_byte;                                 // lds_addr (bytes)
  g0.z = (uint32_t)gaddr;                          // global_addr[31:0]
  g0.w = (uint32_t)(gaddr >> 32) | (2u << 30);     // global_addr[56:32] | type=2 ("image")

  u32x8 g1;
  g1.s0 = (2u << 16)      // data_size = 2 -> 4 bytes
        | (1u << 20)      // pad_enable
        | (3u << 22)      // pad_interval code 3 -> every 16 DWORDs (one 64B row)
        | (3u << 25);     // pad_amount  code 3 -> 4 DWORDs (16B) => row stride 80B = kRS
  g1.s1 = (uint32_t)kT << 16;          // tensor_dim0[15:0] = 2048 (bits 63:48)
  g1.s2 = 8192u << 16;                 // tensor_dim0 hi=0 | tensor_dim1[15:0] = 8192
  g1.s3 = (uint32_t)kTT << 16;         // tensor_dim1 hi=0 | tile_dim0 = 16
  g1.s4 = (uint32_t)kRows;             // tile_dim1 = 64 | tile_dim2 = 0
  g1.s5 = (uint32_t)kT;                // tensor_dim0_stride[31:0] = 2048
  g1.s6 = 0u;                          // stride hi = 0 | tensor_dim1_stride lo = 0 (unused, 2D)
  g1.s7 = 0u;

  asm volatile("tensor_load_to_lds %0, %1"
               :: "s"(g0), "s"(g1)
               : "memory");
}

template <int N>
__device__ __forceinline__ void wait_tensorcnt() {
#if __has_builtin(__builtin_amdgcn_s_wait_tensorcnt)
  __builtin_amdgcn_s_wait_tensorcnt(N);
#else
  asm volatile("s_wait_tensorcnt %0" :: "n"(N) : "memory");
#endif
}

// Depth-optimized EKF step (critical path ~6 dependent ops incl. one v_rcp).
// a, q, scalee (= scale+1e-6), qsc (= q+scalee) are input-only -> off the chain.
__device__ __forceinline__ void ekf_step(float z, float a, float q, float scalee, float qsc,
                                         float& x0, float& x1,
                                         float& P00, float& P01, float& P11) {
  float t01  = fmaf(a, P11, P01);          // d1: P_pred01
  float s2   = P01 + t01;                  // d2
  float x0p  = fmaf(a, x1, x0);            // d1
  float y    = z - x0p;                    // d2
  float P11p = P11 + q;                    // d1
  float P00q = P00 + qsc;                  // d1
  float S    = fmaf(a, s2, P00q);          // d3: P_pred00 + scale + 1e-6
  float Sinv = __builtin_amdgcn_rcpf(S);   // d4 (trans)
  float P00p = S - scalee;                 // d4: P_pred00
  float t01q = t01 * t01;                  // d2
  float w    = Sinv * y;                   // d5
  float om   = scalee * Sinv;              // d5: 1 - K0 == (S - P00p)/S
  x0  = fmaf(P00p, w, x0p);                // d6
  x1  = fmaf(t01,  w, x1);                 // d6
  P00 = om * P00p;                         // d6
  P01 = om * t01;                          // d6
  P11 = fmaf(-t01q, Sinv, P11p);           // d5
}

// Input-only per-step coefficients (off the recurrence critical path).
// rho = 0.5 + 0.5*sigmoid(10h-5) = 0.75 + 0.25*tanh(5h-2.5)  -> single v_tanh_f32 on gfx1250.
__device__ __forceinline__ void ekf_coef(float h, float sg,
                                         float& a, float& q, float& scalee, float& qsc) {
#if __has_builtin(__builtin_amdgcn_tanhf)
  float th    = __builtin_amdgcn_tanhf(fmaf(5.0f, h, -2.5f));
  a           = fmaf(0.25f, th, 0.75f);    // dt * rho, dt = 1
#else
  float e     = __builtin_amdgcn_exp2f((5.0f - 10.0f * h) * 1.44269504088896340736f);
  float sgm   = __builtin_amdgcn_rcpf(1.0f + e);
  a           = fmaf(0.5f, sgm, 0.5f);
#endif
  float scale = fmaxf(100.0f * sg, 1.0f);
  q           = 0.1f * scale;
  scalee      = scale + 1e-6f;
  qsc         = q + scalee;
}

__global__ void __launch_bounds__(32)
ekf_scan(const float* __restrict__ price,
         const float* __restrict__ hurst,
         const float* __restrict__ vsig,
         float* __restrict__ out) {
  __shared__ float lds[2 * kBufSz];

  const int lane = threadIdx.x & 31;
  const int row0 = blockIdx.x * kRows;      // 64 rows per wave; lane owns rows lane, lane+32

  const uint32_t lds_base = (uint32_t)(uintptr_t)(&lds[0]);
  const uint64_t gp = (uint64_t)(uintptr_t)price;
  const uint64_t gh = (uint64_t)(uintptr_t)hurst;
  const uint64_t gs = (uint64_t)(uintptr_t)vsig;
  const uint64_t rowbase = (uint64_t)row0 * kT * sizeof(float);

  auto prefetch_tile = [&](int buf, int t0) {
    const uint32_t lbuf = lds_base + (uint32_t)(buf * kBufSz * sizeof(float));
    const uint64_t goff = rowbase + (uint64_t)t0 * sizeof(float);
    tdm_load_2d(lbuf,                                         gp + goff);
    tdm_load_2d(lbuf + (uint32_t)(kInSz * sizeof(float)),     gh + goff);
    tdm_load_2d(lbuf + (uint32_t)(2 * kInSz * sizeof(float)), gs + goff);
  };

  prefetch_tile(0, 0);

  // Two independent EKF states per lane -> 2-way ILP inside the wave.
  float ax0 = 0.f, ax1 = 0.f, aP00 = 1.f, aP01 = 0.f, aP11 = 1.f;   // row = row0+lane
  float bx0 = 0.f, bx1 = 0.f, bP00 = 1.f, bP01 = 0.f, bP11 = 1.f;   // row = row0+lane+32

  float* outA = out + (size_t)(row0 + lane) * kT * 2;
  float* outB = out + (size_t)(row0 + lane + 32) * kT * 2;

  for (int tile = 0; tile < kTiles; ++tile) {
    const int buf = tile & 1;
    if (tile + 1 < kTiles) {
      prefetch_tile(buf ^ 1, (tile + 1) * kTT);  // overlap next tile's DMA with compute
      wait_tensorcnt<3>();                       // only this tile's 3 DMAs must be done
    } else {
      wait_tensorcnt<0>();
    }

    const float* zrA = &lds[buf * kBufSz + lane * kRS];
    const float* zrB = zrA + 32 * kRS;
    const float* hrA = zrA + kInSz;
    const float* hrB = zrB + kInSz;
    const float* srA = zrA + 2 * kInSz;
    const float* srB = zrB + 2 * kInSz;

    if (tile == 0) {                             // x0 = price[0]; v0 = price[1]-price[0]
      ax0 = zrA[0];  ax1 = zrA[1] - zrA[0];
      bx0 = zrB[0];  bx1 = zrB[1] - zrB[0];
    }

#pragma unroll
    for (int tl = 0; tl < kTT; tl += 2) {
      float aa, aq, ase, aqs, ba, bq, bse, bqs;
      float4 oa, ob;

      ekf_coef(hrA[tl], srA[tl], aa, aq, ase, aqs);
      ekf_coef(hrB[tl], srB[tl], ba, bq, bse, bqs);
      ekf_step(zrA[tl], aa, aq, ase, aqs, ax0, ax1, aP00, aP01, aP11);
      ekf_step(zrB[tl], ba, bq, bse, bqs, bx0, bx1, bP00, bP01, bP11);
      oa.x = ax0; oa.y = ax1;
      ob.x = bx0; ob.y = bx1;

      ekf_coef(hrA[tl + 1], srA[tl + 1], aa, aq, ase, aqs);
      ekf_coef(hrB[tl + 1], srB[tl + 1], ba, bq, bse, bqs);
      ekf_step(zrA[tl + 1], aa, aq, ase, aqs, ax0, ax1, aP00, aP01, aP11);
      ekf_step(zrB[tl + 1], ba, bq, bse, bqs, bx0, bx1, bP00, bP01, bP11);
      oa.z = ax0; oa.w = ax1;
      ob.z = bx0; ob.w = bx1;

      const size_t off = (size_t)(tile * kTT + tl) * 2;
      *reinterpret_cast<float4*>(outA + off) = oa;   // 16B store; L2 write-combines streams
      *reinterpret_cast<float4*>(outB + off) = ob;
    }
  }
}

} // namespace

extern "C" void kernel_launch(void* const* d_in, const int* in_sizes, int n_in,
                              void* d_out, int out_size, void* d_ws, size_t ws_size,
                              hipStream_t stream) {
  const float* price = (const float*)d_in[0];
  const float* hurst = (const float*)d_in[1];
  const float* vsig  = (const float*)d_in[2];
  float* out = (float*)d_out;

  const int rows   = in_sizes[0] / kT;     // B = 8192
  const int blocks = rows / kRows;         // one wave32 per block, 64 rows each -> 128 blocks

  ekf_scan<<<dim3(blocks), dim3(32), 0, stream>>>(price, hurst, vsig, out);
}